// Encoder_22668837388855
// MI455X (gfx1250) — compile-verified
//
#include <hip/hip_runtime.h>

// ---------------------------------------------------------------------------
// Encoder: out[n, 0] = x[n]; out[n, c] = +1 if frac(x*2^(c-1)) < 0.5 else -1,
// c = 1..15.  Pure streaming problem: 16 MB in, 256 MB out => ~11.7 us at
// 23.3 TB/s.  Strategy: compute 32 KB row-tiles into LDS (double buffered),
// then drain each tile with ONE Tensor-Data-Mover DMA (tensor_store_from_lds,
// TENSORcnt) so the memory system sees large linear bursts while the waves
// compute the next tile.
// ---------------------------------------------------------------------------

typedef float    f32x4 __attribute__((ext_vector_type(4)));
typedef unsigned u32x4 __attribute__((ext_vector_type(4)));
typedef int      i32x4 __attribute__((ext_vector_type(4)));
typedef int      i32x8 __attribute__((ext_vector_type(8)));

constexpr unsigned ROWS_PER_TILE    = 512;                       // 512 rows
constexpr unsigned TILE_FLOATS      = ROWS_PER_TILE * 16;        // 8192 floats = 32 KB
constexpr unsigned TILE_QUADS       = TILE_FLOATS / 4;           // 2048 float4
constexpr unsigned BLOCK_THREADS    = 256;                       // 8 waves (wave32)
constexpr unsigned QUADS_PER_THREAD = TILE_QUADS / BLOCK_THREADS;// 8

// One float4 of the output tile.  Quad index e in [0, 2048): row = e>>2,
// column group cg = e&3 covers columns 4cg..4cg+3.
__device__ __forceinline__ f32x4 encode_quad(const float* __restrict__ X,
                                             unsigned rowBase, unsigned e) {
  const unsigned row = e >> 2;
  const unsigned cg  = e & 3u;
  const float    x   = X[rowBase + row];
  // u = floor(x * 2^15): exact (pure exponent shift + truncation), holds
  // fraction bits b1..b15 of x.  square(2*pi*x*2^i) = -1 iff bit b_{i+1} set.
  const unsigned u  = (unsigned)(x * 32768.0f);
  const unsigned c0 = cg << 2;                 // first column of this quad
  f32x4 v;
  const float f0 = __uint_as_float(0x3F800000u | ((u << (16 + c0)) & 0x80000000u));
  v.x = (c0 == 0) ? x : f0;                    // column 0 passes x through
  v.y = __uint_as_float(0x3F800000u | ((u << (17 + c0)) & 0x80000000u));
  v.z = __uint_as_float(0x3F800000u | ((u << (18 + c0)) & 0x80000000u));
  v.w = __uint_as_float(0x3F800000u | ((u << (19 + c0)) & 0x80000000u));
  return v;
}

__global__ __launch_bounds__(BLOCK_THREADS) void encoder_main(
    const float* __restrict__ X, float* __restrict__ OUT, unsigned numTiles) {
  __shared__ f32x4 buf[2][TILE_QUADS];         // 2 x 32 KB, double buffered
  const unsigned t  = threadIdx.x;
  unsigned       it = 0;

  for (unsigned tile = blockIdx.x; tile < numTiles; tile += gridDim.x, ++it) {
    const unsigned b       = it & 1u;
    const unsigned rowBase = tile * ROWS_PER_TILE;

#pragma unroll
    for (unsigned k = 0; k < QUADS_PER_THREAD; ++k) {
      const unsigned e = t + k * BLOCK_THREADS;     // contiguous across lanes
      buf[b][e] = encode_quad(X, rowBase, e);       // ds_store_b128
    }
    __syncthreads();  // all LDS writes of this tile visible

#if defined(__HIP_DEVICE_COMPILE__) && __has_builtin(__builtin_amdgcn_tensor_store_from_lds)
    if ((t >> 5) == 0) {  // wave 0 only (wave-uniform branch; TDM ignores EXEC)
      const unsigned long long gaddr =
          (unsigned long long)(OUT + (size_t)rowBase * 16u);
      const unsigned ldsOff = (unsigned)(unsigned long long)&buf[b][0];

      // D# group 0: count=1 | lds_addr | global_addr[56:0] | type=2
      u32x4 g0 = { 1u,
                   ldsOff,
                   (unsigned)gaddr,
                   (unsigned)((gaddr >> 32) & 0x01FFFFFFu) | 0x80000000u };
      // D# group 1: data_size=2 (4B); tensor_dim0 = tile_dim0 = 8192 (1-D
      // contiguous tile); tensor_dim1 = tile_dim1 = 1; stride0 = 8192.
      i32x8 g1 = { (int)(2u << 16),                        // data_size=4B, mask=0
                   (int)((TILE_FLOATS & 0xFFFFu) << 16),   // tensor_dim0[15:0]
                   (int)(1u << 16),                        // tensor_dim1 = 1
                   (int)((TILE_FLOATS & 0xFFFFu) << 16),   // tile_dim0 = 8192
                   1,                                      // tile_dim1=1, tile_dim2=0
                   (int)TILE_FLOATS,                       // tensor_dim0_stride lo
                   0, 0 };
      i32x4 z4 = (i32x4)0;
#if __clang_major__ >= 23
      i32x8 z8 = (i32x8)0;
      __builtin_amdgcn_tensor_store_from_lds(g0, g1, z4, z4, z8, 0);
#else
      __builtin_amdgcn_tensor_store_from_lds(g0, g1, z4, z4, 0);
#endif
#if __has_builtin(__builtin_amdgcn_s_wait_tensorcnt)
      __builtin_amdgcn_s_wait_tensorcnt(1);   // <=1 outstanding => other buffer free
#else
      asm volatile("s_wait_tensorcnt 0x1" ::: "memory");
#endif
    }
#else
    // Fallback (no TDM builtin): drain the tile with nontemporal b128 stores.
    {
      f32x4* __restrict__ dst = (f32x4*)(OUT + (size_t)rowBase * 16u);
#pragma unroll
      for (unsigned k = 0; k < QUADS_PER_THREAD; ++k) {
        const unsigned e = t + k * BLOCK_THREADS;
        __builtin_nontemporal_store(buf[b][e], dst + e);
      }
    }
#endif
    __syncthreads();  // gate buffer reuse on the DMA issued above
  }

#if defined(__HIP_DEVICE_COMPILE__) && __has_builtin(__builtin_amdgcn_s_wait_tensorcnt)
  __builtin_amdgcn_s_wait_tensorcnt(0);       // drain (S_ENDPGM also waits idle)
#endif
}

// Generic tail for row counts not divisible by ROWS_PER_TILE (N=4M -> unused).
__global__ void encoder_tail(const float* __restrict__ X, float* __restrict__ OUT,
                             unsigned startRow, unsigned nRows) {
  const unsigned idx = blockIdx.x * blockDim.x + threadIdx.x;
  if (idx >= nRows * 16u) return;
  const unsigned row = startRow + (idx >> 4);
  const unsigned c   = idx & 15u;
  const float    x   = X[row];
  const unsigned u   = (unsigned)(x * 32768.0f);
  const float    v   = (c == 0)
      ? x
      : __uint_as_float(0x3F800000u | ((u << (16 + c)) & 0x80000000u));
  OUT[(size_t)row * 16u + c] = v;
}

extern "C" void kernel_launch(void* const* d_in, const int* in_sizes, int n_in,
                              void* d_out, int out_size, void* d_ws, size_t ws_size,
                              hipStream_t stream) {
  (void)n_in; (void)out_size; (void)d_ws; (void)ws_size;
  const float* x   = (const float*)d_in[0];
  float*       out = (float*)d_out;
  const unsigned N        = (unsigned)in_sizes[0];
  const unsigned numTiles = N / ROWS_PER_TILE;
  const unsigned rem      = N % ROWS_PER_TILE;

  if (numTiles) {
    const unsigned blocks = numTiles < 2048u ? numTiles : 2048u;
    encoder_main<<<blocks, BLOCK_THREADS, 0, stream>>>(x, out, numTiles);
  }
  if (rem) {
    const unsigned startRow = numTiles * ROWS_PER_TILE;
    const unsigned total    = rem * 16u;
    encoder_tail<<<(total + 255u) / 256u, 256, 0, stream>>>(x, out, startRow, rem);
  }
}